// PerJointAlignedL1Loss_79345225826861
// MI455X (gfx1250) — compile-verified
//
#include <hip/hip_runtime.h>
#include <cmath>

// ---------------------------------------------------------------------------
// PerJointAlignedL1Loss (Procrustes-aligned L1) for MI455X / gfx1250.
//   pred, gt: [B=1024, N=6890, 3] f32   out: [B, N, 3] f32
// Bandwidth-bound (~423 MB total traffic => ~18us @ 23.3 TB/s).
// Pass 2 (per-batch 3x3 solve) uses V_WMMA_F32_16X16X4_F32 for the 3x3
// matrix products (K^T K, K*V, V'*U^T), zero-padded into the 16x16x4 shape.
// ---------------------------------------------------------------------------

typedef __attribute__((ext_vector_type(2))) float v2f;
typedef __attribute__((ext_vector_type(8))) float v8f;

constexpr int kB = 1024;
constexpr int kN = 6890;
#define EPS_K 1e-8f

// ---------------- Pass 1: per-batch moment reduction -----------------------
// mom[b][0..8]  = sum p_i * g_j   (row-major i*3+j)
// mom[b][9..11] = sum p
// mom[b][12..14]= sum g
// mom[b][15..17]= sum p_i^2
__global__ void __launch_bounds__(256)
k_moments(const float* __restrict__ pred, const float* __restrict__ gt,
          float* __restrict__ mom) {
  const int b = blockIdx.x;
  const float* p = pred + (size_t)b * kN * 3;
  const float* g = gt   + (size_t)b * kN * 3;

  float acc[18];
#pragma unroll
  for (int i = 0; i < 18; ++i) acc[i] = 0.f;

  for (int n = threadIdx.x; n < kN; n += blockDim.x) {
    __builtin_prefetch(p + (n + blockDim.x) * 3, 0, 0);  // global_prefetch_b8
    __builtin_prefetch(g + (n + blockDim.x) * 3, 0, 0);
    float px = p[n * 3 + 0], py = p[n * 3 + 1], pz = p[n * 3 + 2];
    float gx = g[n * 3 + 0], gy = g[n * 3 + 1], gz = g[n * 3 + 2];
    acc[0] += px * gx; acc[1] += px * gy; acc[2] += px * gz;
    acc[3] += py * gx; acc[4] += py * gy; acc[5] += py * gz;
    acc[6] += pz * gx; acc[7] += pz * gy; acc[8] += pz * gz;
    acc[9]  += px; acc[10] += py; acc[11] += pz;
    acc[12] += gx; acc[13] += gy; acc[14] += gz;
    acc[15] += px * px; acc[16] += py * py; acc[17] += pz * pz;
  }

  // wave32 butterfly reduction
#pragma unroll
  for (int i = 0; i < 18; ++i) {
    float v = acc[i];
#pragma unroll
    for (int off = 16; off > 0; off >>= 1) v += __shfl_xor(v, off, 32);
    acc[i] = v;
  }

  __shared__ float sred[8][18];
  const int lane = threadIdx.x & 31, wave = threadIdx.x >> 5;
  if (lane == 0) {
#pragma unroll
    for (int i = 0; i < 18; ++i) sred[wave][i] = acc[i];
  }
  __syncthreads();
  if (threadIdx.x < 18) {
    float s = 0.f;
    const int nw = blockDim.x >> 5;
    for (int w = 0; w < nw; ++w) s += sred[w][threadIdx.x];
    mom[b * 20 + threadIdx.x] = s;
  }
}

// ---------------- WMMA 3x3 helpers (one wave, f32 16x16x4) -----------------
// A fragment (16x4, f32): lanes 0-15 M=0..15 / lanes 16-31 M=0..15;
//   VGPR0 holds K=0 (lo half) / K=2 (hi half), VGPR1 holds K=1 / K=3.
// B fragment (4x16, f32): lane = N column; VGPR0 holds K=0 / K=2,
//   VGPR1 holds K=1 / K=3 (B rows striped across lanes within a VGPR).
// C/D (16x16 f32): VGPR r = row r (lanes 0-15), row 8+r (lanes 16-31).
__device__ inline v2f frag_a3(const float* s, int lane) {
  int m = lane & 15;
  int kb = (lane >> 4) << 1;  // 0 or 2
  v2f a;
  a.x = (m < 3 && kb < 3)     ? s[m * 3 + kb]     : 0.f;
  a.y = (m < 3 && kb + 1 < 3) ? s[m * 3 + kb + 1] : 0.f;
  return a;
}
__device__ inline v2f frag_b3(const float* s, int lane) {
  int n = lane & 15;
  int kb = (lane >> 4) << 1;
  v2f v;
  v.x = (n < 3 && kb < 3)     ? s[kb * 3 + n]       : 0.f;
  v.y = (n < 3 && kb + 1 < 3) ? s[(kb + 1) * 3 + n] : 0.f;
  return v;
}
// sOut = sX(3x3) * sY(3x3), all row-major in LDS
__device__ inline void mm3(const float* sX, const float* sY, float* sOut, int lane) {
  v2f a = frag_a3(sX, lane);
  v2f bb = frag_b3(sY, lane);
  v8f c = {};
  c = __builtin_amdgcn_wmma_f32_16x16x4_f32(false, a, false, bb,
                                            (short)0, c, false, false);
  if (lane < 3) {           // lane n holds column n, rows 0..7 in c[0..7]
    sOut[0 * 3 + lane] = c[0];
    sOut[1 * 3 + lane] = c[1];
    sOut[2 * 3 + lane] = c[2];
  }
}

// ---------------- Pass 2: per-batch Procrustes solve (1 wave/batch) --------
__global__ void __launch_bounds__(32)
k_solve(const float* __restrict__ mom, float* __restrict__ aff) {
  const int b = blockIdx.x;
  const int lane = threadIdx.x;
  const float invN = 1.0f / (float)kN;
  const float* M = mom + b * 20;

  __shared__ float sK[9], sTmp[9], sTmp2[9], sOut[9];

  // centered cross-covariance K = M12 - sp*sg^T/N + EPS (EPS on every entry)
  if (lane < 9) {
    int r = lane / 3, cc = lane % 3;
    sK[lane] = M[lane] - M[9 + r] * M[12 + cc] * invN + EPS_K;
  }
  __syncthreads();

  float K00 = sK[0], K01 = sK[1], K02 = sK[2];
  float K10 = sK[3], K11 = sK[4], K12 = sK[5];
  float K20 = sK[6], K21 = sK[7], K22 = sK[8];

  float sp0 = M[9],  sp1 = M[10], sp2 = M[11];
  float sg0 = M[12], sg1 = M[13], sg2 = M[14];
  float var1 = (M[15] + M[16] + M[17]) - (sp0 * sp0 + sp1 * sp1 + sp2 * sp2) * invN;

  // K^T
  if (lane < 9) sTmp[lane] = sK[(lane % 3) * 3 + (lane / 3)];
  __syncthreads();

  mm3(sTmp, sK, sOut, lane);  // WMMA #1: A = K^T K (symmetric PSD)
  __syncthreads();
  float A00 = sOut[0], A01 = sOut[1], A02 = sOut[2];
  float A11 = sOut[4], A12 = sOut[5], A22 = sOut[8];

  // cyclic Jacobi eigensolve of A; V accumulates eigenvectors (columns)
  float V00 = 1, V01 = 0, V02 = 0, V10 = 0, V11 = 1, V12 = 0, V20 = 0, V21 = 0, V22 = 1;
  auto rot = [](float& app, float& aqq, float& apq, float& arp, float& arq,
                float& vap, float& vaq, float& vbp, float& vbq, float& vcp, float& vcq) {
    float thr = 1e-12f * (fabsf(app) + fabsf(aqq)) + 1e-30f;
    if (fabsf(apq) <= thr) return;
    float tau = (aqq - app) / (2.0f * apq);
    float t = ((tau >= 0.f) ? 1.f : -1.f) / (fabsf(tau) + sqrtf(1.f + tau * tau));
    float c = 1.0f / sqrtf(1.f + t * t);
    float s = t * c;
    app -= t * apq; aqq += t * apq; apq = 0.f;
    float x = arp, y = arq;
    arp = c * x - s * y; arq = s * x + c * y;
    x = vap; y = vaq; vap = c * x - s * y; vaq = s * x + c * y;
    x = vbp; y = vbq; vbp = c * x - s * y; vbq = s * x + c * y;
    x = vcp; y = vcq; vcp = c * x - s * y; vcq = s * x + c * y;
  };
#pragma unroll
  for (int sweep = 0; sweep < 8; ++sweep) {
    rot(A00, A11, A01, A02, A12, V00, V01, V10, V11, V20, V21);  // (p,q)=(0,1)
    rot(A00, A22, A02, A01, A12, V00, V02, V10, V12, V20, V22);  // (p,q)=(0,2)
    rot(A11, A22, A12, A01, A02, V01, V02, V11, V12, V21, V22);  // (p,q)=(1,2)
  }

  // sort eigenpairs descending (SVD convention), swap V columns along
  float e0 = A00, e1 = A11, e2 = A22;
  auto csw = [](float& ea, float& eb, float& x0, float& y0, float& x1, float& y1,
                float& x2, float& y2) {
    if (ea < eb) {
      float t = ea; ea = eb; eb = t;
      t = x0; x0 = y0; y0 = t;
      t = x1; x1 = y1; y1 = t;
      t = x2; x2 = y2; y2 = t;
    }
  };
  csw(e0, e1, V00, V01, V10, V11, V20, V21);
  csw(e0, e2, V00, V02, V10, V12, V20, V22);
  csw(e1, e2, V01, V02, V11, V12, V21, V22);

  float is0 = 1.f / fmaxf(sqrtf(fmaxf(e0, 0.f)), 1e-20f);
  float is1 = 1.f / fmaxf(sqrtf(fmaxf(e1, 0.f)), 1e-20f);
  float is2 = 1.f / fmaxf(sqrtf(fmaxf(e2, 0.f)), 1e-20f);

  if (lane == 0) {
    sTmp[0] = V00; sTmp[1] = V01; sTmp[2] = V02;
    sTmp[3] = V10; sTmp[4] = V11; sTmp[5] = V12;
    sTmp[6] = V20; sTmp[7] = V21; sTmp[8] = V22;
  }
  __syncthreads();
  mm3(sK, sTmp, sOut, lane);  // WMMA #2: T = K * V  -> U = T * S^-1
  __syncthreads();
  float U00 = sOut[0] * is0, U01 = sOut[1] * is1, U02 = sOut[2] * is2;
  float U10 = sOut[3] * is0, U11 = sOut[4] * is1, U12 = sOut[5] * is2;
  float U20 = sOut[6] * is0, U21 = sOut[7] * is1, U22 = sOut[8] * is2;

  // sign(det(U V^T)) == sign(det K) since singular values >= 0
  float detK = K00 * (K11 * K22 - K12 * K21) - K01 * (K10 * K22 - K12 * K20) +
               K02 * (K10 * K21 - K11 * K20);
  float sgn = (detK >= 0.f) ? 1.f : -1.f;

  __syncthreads();
  if (lane == 0) {
    // V' = V * diag(1,1,sgn)
    sTmp[0] = V00; sTmp[1] = V01; sTmp[2] = V02 * sgn;
    sTmp[3] = V10; sTmp[4] = V11; sTmp[5] = V12 * sgn;
    sTmp[6] = V20; sTmp[7] = V21; sTmp[8] = V22 * sgn;
    // U^T
    sTmp2[0] = U00; sTmp2[1] = U10; sTmp2[2] = U20;
    sTmp2[3] = U01; sTmp2[4] = U11; sTmp2[5] = U21;
    sTmp2[6] = U02; sTmp2[7] = U12; sTmp2[8] = U22;
  }
  __syncthreads();
  mm3(sTmp, sTmp2, sOut, lane);  // WMMA #3: R = V' * U^T
  __syncthreads();
  float R00 = sOut[0], R01 = sOut[1], R02 = sOut[2];
  float R10 = sOut[3], R11 = sOut[4], R12 = sOut[5];
  float R20 = sOut[6], R21 = sOut[7], R22 = sOut[8];

  // scale = trace(R K) / var1
  float trRK = R00 * K00 + R01 * K10 + R02 * K20 +
               R10 * K01 + R11 * K11 + R12 * K21 +
               R20 * K02 + R21 * K12 + R22 * K22;
  float scale = trRK / var1;

  float m10 = sp0 * invN, m11 = sp1 * invN, m12 = sp2 * invN;  // mu1
  float m20 = sg0 * invN, m21 = sg1 * invN, m22 = sg2 * invN;  // mu2
  float t0 = m20 - scale * (R00 * m10 + R01 * m11 + R02 * m12);
  float t1 = m21 - scale * (R10 * m10 + R11 * m11 + R12 * m12);
  float t2 = m22 - scale * (R20 * m10 + R21 * m11 + R22 * m12);

  if (lane == 0) {
    float* A = aff + b * 12;
    A[0] = scale * R00; A[1] = scale * R01; A[2] = scale * R02;
    A[3] = scale * R10; A[4] = scale * R11; A[5] = scale * R12;
    A[6] = scale * R20; A[7] = scale * R21; A[8] = scale * R22;
    A[9] = t0; A[10] = t1; A[11] = t2;
  }
}

// ---------------- Pass 3: apply affine + L1 (streaming) --------------------
__global__ void __launch_bounds__(256)
k_apply(const float* __restrict__ pred, const float* __restrict__ gt,
        const float* __restrict__ aff, float* __restrict__ out) {
  const int b = blockIdx.y;
  const int n = blockIdx.x * blockDim.x + threadIdx.x;
  if (n >= kN) return;
  const float* A = aff + b * 12;  // uniform per block -> scalar loads
  float a00 = A[0], a01 = A[1], a02 = A[2];
  float a10 = A[3], a11 = A[4], a12 = A[5];
  float a20 = A[6], a21 = A[7], a22 = A[8];
  float t0 = A[9], t1 = A[10], t2 = A[11];

  size_t base = ((size_t)b * kN + n) * 3;
  float px = pred[base], py = pred[base + 1], pz = pred[base + 2];
  float gx = gt[base],   gy = gt[base + 1],   gz = gt[base + 2];

  out[base]     = fabsf(a00 * px + a01 * py + a02 * pz + t0 - gx);
  out[base + 1] = fabsf(a10 * px + a11 * py + a12 * pz + t1 - gy);
  out[base + 2] = fabsf(a20 * px + a21 * py + a22 * pz + t2 - gz);
}

// ---------------------------------------------------------------------------
extern "C" void kernel_launch(void* const* d_in, const int* in_sizes, int n_in,
                              void* d_out, int out_size, void* d_ws, size_t ws_size,
                              hipStream_t stream) {
  (void)in_sizes; (void)n_in; (void)out_size; (void)ws_size;
  const float* pred = (const float*)d_in[0];
  const float* gt   = (const float*)d_in[1];
  float* out = (float*)d_out;

  float* mom = (float*)d_ws;          // [kB][20]
  float* aff = mom + (size_t)kB * 20; // [kB][12]

  k_moments<<<kB, 256, 0, stream>>>(pred, gt, mom);
  k_solve<<<kB, 32, 0, stream>>>(mom, aff);
  dim3 g3((kN + 255) / 256, kB);
  k_apply<<<g3, 256, 0, stream>>>(pred, gt, aff, out);
}